// SegmentLUT_83021717831949
// MI455X (gfx1250) — compile-verified
//
#include <hip/hip_runtime.h>

#define NSEG   8
#define BLOCK  256
#define PIPE   4      // async stages in flight per wave
#define SLOTS  8      // 2*PIPE LDS ring slots -> no DS-vs-async WAR hazard

typedef float f32x4 __attribute__((ext_vector_type(4)));
typedef int   v4i   __attribute__((ext_vector_type(4)));
typedef __attribute__((address_space(1))) v4i* gv4i_ptr;   // global int4*
typedef __attribute__((address_space(3))) v4i* lv4i_ptr;   // LDS int4*

__device__ __forceinline__ int imax(int a, int b) { return a > b ? a : b; }
__device__ __forceinline__ int imin(int a, int b) { return a < b ? a : b; }

// ---- CDNA5 async global->LDS (ASYNCcnt-tracked) -----------------------------
__device__ __forceinline__ void async_load_f4(const float* __restrict__ gsrc,
                                              f32x4* lds_dst) {
#if __has_builtin(__builtin_amdgcn_global_load_async_to_lds_b128)
  __builtin_amdgcn_global_load_async_to_lds_b128(
      (gv4i_ptr)(v4i*)gsrc,      // param0: v4i32 addrspace(1)* (non-const)
      (lv4i_ptr)(v4i*)lds_dst,   // param1: LDS destination
      0, 0);                     // offset, cpol
#else
  unsigned lds_off = (unsigned)(unsigned long long)lds_dst;  // low 32 bits = LDS byte addr
  asm volatile("global_load_async_to_lds_b128 %0, %1, off"
               :: "v"(lds_off), "v"(gsrc)
               : "memory");
#endif
}

#if __has_builtin(__builtin_amdgcn_s_wait_asynccnt)
#define WAIT_ASYNCCNT(n) __builtin_amdgcn_s_wait_asynccnt(n)
#else
#define WAIT_ASYNCCNT(n) asm volatile("s_wait_asynccnt %0" :: "i"(n))
#endif
#define CFENCE() asm volatile("" ::: "memory")

// ---- per-element fixed-point segment LUT ------------------------------------
// KR[j] = (k & 0xFFFF) | (rs << 16);  BB[j] = (b << ls) + (rs>0 ? 1<<(rs-1) : 0)
__device__ __forceinline__ float seg_eval(float xf, float inv_in, float osc,
                                          const int (&SM)[NSEG],
                                          const int (&KR)[NSEG],
                                          const int (&BB)[NSEG]) {
  float r = __builtin_rintf(xf * inv_in);          // v_mul + v_rndne (RNE = jnp.round)
  int xi = (int)r;                                  // v_cvt_i32_f32
  xi = imin(imax(xi, -32768), 32767);               // v_med3_i32
  int kr = KR[NSEG - 1];
  int bb = BB[NSEG - 1];
#pragma unroll
  for (int j = NSEG - 2; j >= 0; --j) {             // first j with SM[j] >= xi
    bool c = (xi <= SM[j]);                         // 7x v_cmp + 2 cndmask (SGPR srcs)
    kr = c ? KR[j] : kr;
    bb = c ? BB[j] : bb;
  }
  int rs = (int)(((unsigned)kr) >> 16);             // v_lshrrev
  // xi and k are exact int16 values -> eligible for v_mad_i32_i16
  int acc = (int)(short)xi * (int)(short)kr + bb;
  int q = acc >> rs;                                // arithmetic shift (matches jnp)
  q = imin(imax(q, -32768), 32767);                 // v_med3_i32
  return (float)q * osc;                            // v_cvt + v_mul
}

template <bool EXACT>   // EXACT: nf4 % BLOCK == 0 -> no per-lane bounds work
__global__ __launch_bounds__(BLOCK) void seglut_async_kernel(
    const float* __restrict__ x,
    const float* __restrict__ in_scale,
    const float* __restrict__ out_scale,
    const int* __restrict__ alpha,
    const int* __restrict__ beta,
    const int* __restrict__ lshift,
    const int* __restrict__ rshift,
    const int* __restrict__ segmax,
    float* __restrict__ out,
    long long n)
{
  __shared__ f32x4 stage[SLOTS][BLOCK];   // 8 * 256 * 16B = 32 KB

  // Uniform per-segment constants (scalar loads -> SGPRs / SALU folding)
  const float inv_in = 1.0f / in_scale[0];
  const float osc    = out_scale[0];
  int SM[NSEG], KR[NSEG], BB[NSEG];
#pragma unroll
  for (int j = 0; j < NSEG; ++j) {
    int k  = alpha[j];
    int b  = beta[j];
    int ls = lshift[j];
    int rs = rshift[j];
    int rb = (rs > 0) ? (1 << (rs - 1)) : 0;
    BB[j] = (b << ls) + rb;
    KR[j] = (k & 0xFFFF) | (rs << 16);
    SM[j] = segmax[j];
  }

  const long long nf4         = n >> 2;
  const long long totalStages = (nf4 + BLOCK - 1) / BLOCK;
  const int       tid         = threadIdx.x;
  const long long grid        = gridDim.x;

  if (totalStages > 0) {
    // Prologue: fill the pipeline PIPE deep. Clamps are uniform (SALU-only).
#pragma unroll
    for (int p = 0; p < PIPE; ++p) {
      long long st = (long long)blockIdx.x + (long long)p * grid;
      if (st >= totalStages) st = totalStages - 1;
      long long f4 = st * BLOCK + tid;
      if (!EXACT) { if (f4 >= nf4) f4 = nf4 - 1; }
      async_load_f4(x + f4 * 4, &stage[p][tid]);
    }

    long long q = 0;
    for (long long it = blockIdx.x; it < totalStages; it += grid, ++q) {
      // Always exactly PIPE outstanding -> cnt<=PIPE-1 means oldest landed.
      WAIT_ASYNCCNT(PIPE - 1);
      CFENCE();
      const int slot = (int)(q & (SLOTS - 1));
      f32x4 v = stage[slot][tid];                    // ds_load_b128

      // Keep the pipeline full: issue stage it+PIPE*grid into the ring slot
      // PIPE ahead (read 4 iterations ago -> its ds_load already retired).
      {
        long long st = it + (long long)PIPE * grid;
        if (st >= totalStages) st = totalStages - 1;  // uniform SALU clamp
        long long f4 = st * BLOCK + tid;
        if (!EXACT) { if (f4 >= nf4) f4 = nf4 - 1; }
        async_load_f4(x + f4 * 4, &stage[(slot + PIPE) & (SLOTS - 1)][tid]);
      }

      long long f4 = it * BLOCK + tid;
      if (EXACT || f4 < nf4) {
        f32x4 r;
        r.x = seg_eval(v.x, inv_in, osc, SM, KR, BB);
        r.y = seg_eval(v.y, inv_in, osc, SM, KR, BB);
        r.z = seg_eval(v.z, inv_in, osc, SM, KR, BB);
        r.w = seg_eval(v.w, inv_in, osc, SM, KR, BB);
        __builtin_nontemporal_store(r, (f32x4*)(out + f4 * 4));  // write-once stream
      }
    }
  }

  // Scalar tail for n % 4 (n is 2^25 here, but stay general).
  if (blockIdx.x == 0 && tid == 0) {
    for (long long e = nf4 * 4; e < n; ++e) {
      out[e] = seg_eval(x[e], inv_in, osc, SM, KR, BB);
    }
  }
}

extern "C" void kernel_launch(void* const* d_in, const int* in_sizes, int n_in,
                              void* d_out, int out_size, void* d_ws, size_t ws_size,
                              hipStream_t stream) {
  const float* x     = (const float*)d_in[0];
  const float* iscl  = (const float*)d_in[1];
  const float* oscl  = (const float*)d_in[2];
  const int*   alpha = (const int*)d_in[3];
  const int*   beta  = (const int*)d_in[4];
  const int*   lsh   = (const int*)d_in[5];
  const int*   rsh   = (const int*)d_in[6];
  const int*   smax  = (const int*)d_in[7];
  float*       out   = (float*)d_out;

  long long n           = (long long)in_sizes[0];
  long long nf4         = n >> 2;
  long long totalStages = (nf4 + BLOCK - 1) / BLOCK;
  long long blocks      = totalStages < 2048 ? (totalStages > 0 ? totalStages : 1)
                                             : 2048;

  if (nf4 % BLOCK == 0) {
    seglut_async_kernel<true><<<(int)blocks, BLOCK, 0, stream>>>(
        x, iscl, oscl, alpha, beta, lsh, rsh, smax, out, n);
  } else {
    seglut_async_kernel<false><<<(int)blocks, BLOCK, 0, stream>>>(
        x, iscl, oscl, alpha, beta, lsh, rsh, smax, out, n);
  }
}